// ASAP_63428077027632
// MI455X (gfx1250) — compile-verified
//
#include <hip/hip_runtime.h>
#include <hip/hip_bf16.h>
#include <math.h>

// ---------------- problem constants (match reference) ----------------
#define NN      131072      // total nodes
#define NPG     2048        // nodes per graph
#define NB      64          // graphs
#define DEG     8           // in-degree
#define HH      64          // hidden
#define NCLS    10
#define NLAY    9
#define NEGF    (-1e30f)
#define HSTR    66          // padded LDS row stride (floats) for 64-wide rows

#define EDGE_BLOCKS 2048    // persistent blocks; 16384 tiles / 2048 = 8 iters
#define BASE_BLOCKS 512     // 2048 groups / 512 = 4 iters

typedef float v2f __attribute__((ext_vector_type(2)));
typedef float v8f __attribute__((ext_vector_type(8)));
typedef int   v4i __attribute__((vector_size(4 * sizeof(int))));  // matches builtin param

// ---------------------------------------------------------------------
// Async global->LDS weight staging (CDNA5 GLOBAL_LOAD_ASYNC_TO_LDS_B128,
// ASYNCcnt-tracked). Guarded: falls back to plain LDS copy if the builtin
// is not declared by this toolchain. Signature (confirmed by compiler
// diagnostic): (AS1 v4i* src, AS3 dst, imm offset, imm cpol).
// ---------------------------------------------------------------------
#if defined(__AMDGCN__) && __has_builtin(__builtin_amdgcn_global_load_async_to_lds_b128) && \
    __has_builtin(__builtin_amdgcn_s_wait_asynccnt)
#define USE_ASYNC_LDS 1
#endif

__device__ __forceinline__ void stage_weights(const float* __restrict__ g, float* l,
                                              int count, int tid) {
#ifdef USE_ASYNC_LDS
  // one b128 (4 floats) per lane per issue; generic->AS casts via integer:
  // low 32 bits of a generic LDS address are the wave-relative LDS offset.
  for (int idx = tid * 4; idx < count; idx += 128 * 4)
    __builtin_amdgcn_global_load_async_to_lds_b128(
        (__attribute__((address_space(1))) v4i*)(unsigned long long)(g + idx),
        (__attribute__((address_space(3))) v4i*)(unsigned int)(unsigned long long)(l + idx),
        0, 0);
#else
  for (int idx = tid; idx < count; idx += 128) l[idx] = g[idx];
#endif
}

__device__ __forceinline__ void stage_wait() {
#ifdef USE_ASYNC_LDS
  __builtin_amdgcn_s_wait_asynccnt(0);
#endif
}

// =====================================================================
// WMMA helper: C(16x64) += A(16x64 in LDS, ld=HSTR) * B(64x64 row-major in LDS)
// Fragment layouts per CDNA5 ISA 7.12.2:
//  A 16x4 f32 : lanes 0-15 -> M=lane, K={0,1} in vgpr{0,1}; lanes 16-31 -> K={2,3}
//  B 4x16 f32 : lanes 0-15 -> N=lane, K={0,1} in vgpr{0,1}; lanes 16-31 -> K={2,3}
//  C 16x16    : vgpr r -> M=r (lanes 0-15) / M=r+8 (lanes 16-31), N=lane&15
// =====================================================================
__device__ __forceinline__ void wmma_gemm64(const float* hh, const float* W,
                                            int lane, v8f acc[4]) {
  const int half = lane >> 4;
  const int el   = lane & 15;
#pragma unroll
  for (int k0 = 0; k0 < 64; k0 += 4) {
    const int ka = k0 + 2 * half;
    v2f a; a.x = hh[el * HSTR + ka]; a.y = hh[el * HSTR + ka + 1];
#pragma unroll
    for (int nt = 0; nt < 4; ++nt) {
      const int col = nt * 16 + el;
      v2f bv; bv.x = W[ka * 64 + col]; bv.y = W[(ka + 1) * 64 + col];
      acc[nt] = __builtin_amdgcn_wmma_f32_16x16x4_f32(
          false, a, false, bv, (short)0, acc[nt], false, false);
    }
  }
}

__device__ __forceinline__ void store_tile_relu(float* hh, int lane, const v8f acc[4]) {
  const int half = lane >> 4;
  const int el   = lane & 15;
#pragma unroll
  for (int nt = 0; nt < 4; ++nt)
#pragma unroll
    for (int v = 0; v < 8; ++v) {
      const int m = v + 8 * half;
      hh[m * HSTR + nt * 16 + el] = fmaxf(acc[nt][v], 0.f);
    }
}

// =====================================================================
// init alive
// =====================================================================
__global__ void init_alive_kernel(int* alive) {
  int n = blockIdx.x * 256 + threadIdx.x;
  if (n < NN) alive[n] = 1;
}

// =====================================================================
// PNAConv (deg==8 for every node -> scalers are identity: scl=[aggr,aggr,aggr])
// one thread per node
// =====================================================================
__global__ void pna_kernel(const float* __restrict__ x, const float* __restrict__ pos,
                           const float* __restrict__ eattr,
                           const float* __restrict__ pew, const float* __restrict__ peb,
                           const float* __restrict__ prw, const float* __restrict__ prb,
                           const float* __restrict__ ppw, const float* __restrict__ ppb,
                           const float* __restrict__ plw, const float* __restrict__ plb,
                           const int* __restrict__ srcIdx, float* __restrict__ xcur) {
  int n = blockIdx.x * blockDim.x + threadIdx.x;
  if (n >= NN) return;
  float xi[5];
  xi[0] = x[n * 2]; xi[1] = x[n * 2 + 1];
  xi[2] = pos[n * 3]; xi[3] = pos[n * 3 + 1]; xi[4] = pos[n * 3 + 2];
  float s[5], s2[5], mx[5], mnn[5];
#pragma unroll
  for (int j = 0; j < 5; ++j) { s[j] = 0.f; s2[j] = 0.f; mx[j] = -3.4e38f; mnn[j] = -3.4e38f; }
  for (int r = 0; r < DEG; ++r) {
    int e = n * DEG + r;
    int sn = srcIdx[e];
    float xj[5];
    xj[0] = x[sn * 2]; xj[1] = x[sn * 2 + 1];
    xj[2] = pos[sn * 3]; xj[3] = pos[sn * 3 + 1]; xj[4] = pos[sn * 3 + 2];
    float ea = eattr[e];
    float e5[5];
#pragma unroll
    for (int t = 0; t < 5; ++t) e5[t] = ea * pew[t] + peb[t];
#pragma unroll
    for (int j = 0; j < 5; ++j) {
      float m = prb[j];
#pragma unroll
      for (int t = 0; t < 5; ++t)
        m += xi[t] * prw[t * 5 + j] + xj[t] * prw[(5 + t) * 5 + j] + e5[t] * prw[(10 + t) * 5 + j];
      s[j] += m; s2[j] += m * m;
      mx[j] = fmaxf(mx[j], m); mnn[j] = fmaxf(mnn[j], -m);
    }
  }
  float aggr[20];
#pragma unroll
  for (int j = 0; j < 5; ++j) {
    float mean = s[j] * 0.125f;
    float var = fmaxf(s2[j] * 0.125f - mean * mean, 0.f);
    aggr[j] = mean; aggr[5 + j] = -mnn[j]; aggr[10 + j] = mx[j];
    aggr[15 + j] = sqrtf(var + 1e-5f);
  }
  float h[HH];
#pragma unroll
  for (int f = 0; f < HH; ++f) {
    float a = ppb[f];
#pragma unroll
    for (int t = 0; t < 5; ++t) a += xi[t] * ppw[t * HH + f];
#pragma unroll
    for (int q = 0; q < 20; ++q)
      a += aggr[q] * (ppw[(5 + q) * HH + f] + ppw[(25 + q) * HH + f] + ppw[(45 + q) * HH + f]);
    h[f] = a;
  }
  for (int f = 0; f < HH; ++f) {
    float a = plb[f];
#pragma unroll
    for (int j = 0; j < HH; ++j) a += h[j] * plw[j * HH + f];
    xcur[(size_t)n * HH + f] = fmaxf(a, 0.f);
  }
}

// =====================================================================
// base[n] = b0 + pos[n]@(W0[0:3]-W0[3:6]) + xcur[n]@W0[6:70]   (WMMA)
// persistent: 512 blocks grid-stride over 2048 groups of 64 nodes;
// weights staged to LDS once per block.
// =====================================================================
__global__ __launch_bounds__(128) void base_kernel(const float* __restrict__ xcur,
                                                   const float* __restrict__ pos,
                                                   const float* __restrict__ w0,  // [70][64]
                                                   const float* __restrict__ b0,
                                                   float* __restrict__ baseOut) {
  __shared__ __attribute__((aligned(16))) float Wm[64 * 64];
  __shared__ __attribute__((aligned(16))) float b0s[64];
  __shared__ float Wd[3 * 64];
  __shared__ float hhS[4][16 * HSTR];
  __shared__ float posS[4][48];
  const int tid = threadIdx.x, w = tid >> 5, lane = tid & 31;
  const int half = lane >> 4, el = lane & 15;
  stage_weights(w0 + 6 * 64, Wm, 4096, tid);
  stage_weights(b0, b0s, 64, tid);
  for (int idx = tid; idx < 192; idx += 128) Wd[idx] = w0[idx] - w0[192 + idx];
  stage_wait();
  __syncthreads();

  for (int t = blockIdx.x; t < NN / 64; t += BASE_BLOCKS) {
    const int nbase = t * 64 + w * 16;
    {
      const float* xr = xcur + (size_t)(nbase + el) * HH + half * 32;
      float* hr = &hhS[w][el * HSTR + half * 32];
#pragma unroll
      for (int j = 0; j < 32; ++j) hr[j] = xr[j];
      if (half == 0) {
        posS[w][el * 3 + 0] = pos[(size_t)(nbase + el) * 3 + 0];
        posS[w][el * 3 + 1] = pos[(size_t)(nbase + el) * 3 + 1];
        posS[w][el * 3 + 2] = pos[(size_t)(nbase + el) * 3 + 2];
      }
    }
    __syncthreads();
    v8f acc[4];
#pragma unroll
    for (int nt = 0; nt < 4; ++nt) {
      const int col = nt * 16 + el;
      const float bb = b0s[col];
#pragma unroll
      for (int v = 0; v < 8; ++v) {
        const int m = v + 8 * half;
        acc[nt][v] = bb + posS[w][m * 3 + 0] * Wd[col] + posS[w][m * 3 + 1] * Wd[64 + col] +
                     posS[w][m * 3 + 2] * Wd[128 + col];
      }
    }
    wmma_gemm64(&hhS[w][0], Wm, lane, acc);
#pragma unroll
    for (int nt = 0; nt < 4; ++nt)
#pragma unroll
      for (int v = 0; v < 8; ++v) {
        const int m = v + 8 * half;
        baseOut[(size_t)(nbase + m) * HH + nt * 16 + el] = acc[nt][v];
      }
    if (t + BASE_BLOCKS < NN / 64)
      __builtin_prefetch(xcur + (size_t)(t + BASE_BLOCKS) * 64 * HH, 0, 0);
    __syncthreads();   // hhS/posS reused next iteration
  }
}

// =====================================================================
// Edge MLP + masked segment-max (WMMA). wave = 2 nodes = 16 edges.
// hh0 = relu(base[dst] + pos[src]@W0[3:6]); hh1 = relu(hh0@W1+b1);
// hh2 = relu(hh1@W2+b2); xcur_new = relu(max over alive edges) * alive[dst]
// persistent: 2048 blocks grid-stride over 16384 tiles of 8 nodes;
// W1/W2 staged to LDS once per block (async if available).
// =====================================================================
__global__ __launch_bounds__(128) void edge_kernel(const float* __restrict__ base,
                                                   const float* __restrict__ pos,
                                                   const int* __restrict__ srcIdx,
                                                   const int* __restrict__ alive,
                                                   const float* __restrict__ w0,  // [70][64]
                                                   const float* __restrict__ b1v,
                                                   const float* __restrict__ w1,
                                                   const float* __restrict__ b2v,
                                                   const float* __restrict__ w2,
                                                   float* __restrict__ xcurOut) {
  __shared__ __attribute__((aligned(16))) float W1s[64 * 64];
  __shared__ __attribute__((aligned(16))) float W2s[64 * 64];
  __shared__ __attribute__((aligned(16))) float W0e[3 * 64];
  __shared__ __attribute__((aligned(16))) float b1s[64];
  __shared__ __attribute__((aligned(16))) float b2s[64];
  __shared__ float hhS[4][16 * HSTR];
  __shared__ int aliveS[4][16];
  const int tid = threadIdx.x, w = tid >> 5, lane = tid & 31;
  const int half = lane >> 4, el = lane & 15;
  stage_weights(w1, W1s, 4096, tid);
  stage_weights(w2, W2s, 4096, tid);
  stage_weights(w0 + 3 * 64, W0e, 192, tid);
  stage_weights(b1v, b1s, 64, tid);
  stage_weights(b2v, b2s, 64, tid);
  stage_wait();
  __syncthreads();

  for (int t = blockIdx.x; t < NN / 8; t += EDGE_BLOCKS) {
    const int n0 = t * 8 + w * 2;               // first node of this wave
    const int node = n0 + (el >> 3);            // dst node of edge `el`
    const int e = n0 * DEG + el;
    const int sn = srcIdx[e];
    const int eal = alive[node] & alive[sn];
    if (half == 0) aliveS[w][el] = eal;
    const float p0 = pos[(size_t)sn * 3], p1 = pos[(size_t)sn * 3 + 1],
                p2 = pos[(size_t)sn * 3 + 2];
    {
      const float* brow = base + (size_t)node * HH + half * 32;
      float* hrow = &hhS[w][el * HSTR + half * 32];
      const float* w0r = &W0e[half * 32];
#pragma unroll
      for (int j = 0; j < 32; ++j) {
        float v = brow[j] + p0 * w0r[j] + p1 * w0r[64 + j] + p2 * w0r[128 + j];
        hrow[j] = fmaxf(v, 0.f);
      }
    }
    __syncthreads();

    v8f acc[4];
#pragma unroll
    for (int nt = 0; nt < 4; ++nt) {
      const float bb = b1s[nt * 16 + el];
#pragma unroll
      for (int v = 0; v < 8; ++v) acc[nt][v] = bb;
    }
    wmma_gemm64(&hhS[w][0], W1s, lane, acc);
    __syncthreads();
    store_tile_relu(&hhS[w][0], lane, acc);
    __syncthreads();
#pragma unroll
    for (int nt = 0; nt < 4; ++nt) {
      const float bb = b2s[nt * 16 + el];
#pragma unroll
      for (int v = 0; v < 8; ++v) acc[nt][v] = bb;
    }
    wmma_gemm64(&hhS[w][0], W2s, lane, acc);
    __syncthreads();
    store_tile_relu(&hhS[w][0], lane, acc);
    __syncthreads();

    // reduction: lane-half picks node, el picks feature column (4 feats / lane)
    const int myNode = n0 + half;
    const int aD = alive[myNode];
#pragma unroll
    for (int ff = 0; ff < 4; ++ff) {
      const int f = el + ff * 16;
      float m = NEGF;
#pragma unroll
      for (int r = 0; r < DEG; ++r)
        if (aliveS[w][half * 8 + r]) m = fmaxf(m, hhS[w][(half * 8 + r) * HSTR + f]);
      xcurOut[(size_t)myNode * HH + f] = aD ? fmaxf(m, 0.f) : 0.f;
    }
    if (t + EDGE_BLOCKS < NN / 8)
      __builtin_prefetch(base + (size_t)(t + EDGE_BLOCKS) * 8 * HH, 0, 0);
    __syncthreads();   // hhS/aliveS reused next iteration
  }
}

// =====================================================================
// graph mean pooling over alive nodes -> xs slot
// =====================================================================
__global__ void gmp_kernel(const float* __restrict__ xcur, const int* __restrict__ alive,
                           float* __restrict__ xsSlot) {
  __shared__ float partial[4][64];
  __shared__ float cnt[4];
  const int b = blockIdx.x, tid = threadIdx.x;
  const int f = tid & 63, g = tid >> 6;
  float acc = 0.f, c = 0.f;
  for (int r = g; r < NPG; r += 4) {
    const int n = b * NPG + r;
    if (alive[n]) { acc += xcur[(size_t)n * HH + f]; c += 1.f; }
  }
  partial[g][f] = acc;
  if (f == 0) cnt[g] = c;
  __syncthreads();
  if (g == 0) {
    float ssum = partial[0][f] + partial[1][f] + partial[2][f] + partial[3][f];
    float cc = cnt[0] + cnt[1] + cnt[2] + cnt[3];
    xsSlot[b * HH + f] = ssum / cc;
  }
}

// =====================================================================
// pool stage 1: mcl = max(xcur, max_{alive edges} xcur[src]); q = mcl@plw+plb
// store qdot = q.att[:64], xdot = xcur.att[64:], s_s = lrelu(qdot+xdot)
// =====================================================================
__global__ void pool_q_kernel(const float* __restrict__ xcur, const int* __restrict__ srcIdx,
                              const int* __restrict__ alive, const float* __restrict__ plw,
                              const float* __restrict__ plb, const float* __restrict__ att,
                              float* __restrict__ qdot, float* __restrict__ xdot,
                              float* __restrict__ s_s) {
  int n = blockIdx.x * blockDim.x + threadIdx.x;
  if (n >= NN) return;
  float mcl[HH];
#pragma unroll
  for (int f = 0; f < HH; ++f) mcl[f] = xcur[(size_t)n * HH + f];
  float xd = 0.f;
#pragma unroll
  for (int f = 0; f < HH; ++f) xd += mcl[f] * att[64 + f];
  const int aD = alive[n];
  for (int r = 0; r < DEG; ++r) {
    const int sn = srcIdx[n * DEG + r];
    if (aD && alive[sn]) {
#pragma unroll
      for (int f = 0; f < HH; ++f) mcl[f] = fmaxf(mcl[f], xcur[(size_t)sn * HH + f]);
    }
  }
  float qd = 0.f;
  for (int f = 0; f < HH; ++f) {
    float q = plb[f];
#pragma unroll
    for (int k = 0; k < HH; ++k) q += mcl[k] * plw[k * HH + f];
    qd += q * att[f];
  }
  float ss = qd + xd; ss = ss > 0.f ? ss : 0.2f * ss;
  qdot[n] = qd; xdot[n] = xd; s_s[n] = ss;
}

// =====================================================================
// pool stage 2: attention-softmax cluster rep xc + LEConv partials
// =====================================================================
__global__ void pool_xc_kernel(const float* __restrict__ xcur, const int* __restrict__ srcIdx,
                               const int* __restrict__ alive, const float* __restrict__ qdotA,
                               const float* __restrict__ xdotA, const float* __restrict__ ssA,
                               const float* __restrict__ lw1, const float* __restrict__ lw2,
                               const float* __restrict__ lw3, const float* __restrict__ lb,
                               float* __restrict__ xc, float* __restrict__ w3dot,
                               float* __restrict__ lecpart) {
  int n = blockIdx.x * blockDim.x + threadIdx.x;
  if (n >= NN) return;
  const int aD = alive[n];
  const float qd = qdotA[n], ss = ssA[n];
  float smx = ss;
  int sns[DEG], eal[DEG];
  float se[DEG];
  int ecnt = 0;
#pragma unroll
  for (int r = 0; r < DEG; ++r) {
    const int sn = srcIdx[n * DEG + r];
    sns[r] = sn;
    const int a = aD && alive[sn];
    eal[r] = a;
    float v = qd + xdotA[sn];
    v = v > 0.f ? v : 0.2f * v;
    se[r] = v;
    if (a) { smx = fmaxf(smx, v); ecnt++; }
  }
  const float es = expf(ss - smx);
  float den = es;
  float wgt[DEG];
#pragma unroll
  for (int r = 0; r < DEG; ++r) {
    wgt[r] = eal[r] ? expf(se[r] - smx) : 0.f;
    den += wgt[r];
  }
  float acc[HH];
#pragma unroll
  for (int f = 0; f < HH; ++f) acc[f] = es * xcur[(size_t)n * HH + f];
#pragma unroll
  for (int r = 0; r < DEG; ++r) {
    if (eal[r]) {
      const int sn = sns[r];
#pragma unroll
      for (int f = 0; f < HH; ++f) acc[f] += wgt[r] * xcur[(size_t)sn * HH + f];
    }
  }
  const float inv = 1.f / den;
  float l1 = 0.f, l2 = 0.f, l3 = 0.f;
#pragma unroll
  for (int f = 0; f < HH; ++f) {
    const float v = acc[f] * inv;
    xc[(size_t)n * HH + f] = v;
    l1 += v * lw1[f]; l2 += v * lw2[f]; l3 += v * lw3[f];
  }
  w3dot[n] = l3;
  lecpart[n] = l1 + (float)ecnt * l2 + lb[0];
}

// =====================================================================
// pool stage 3: LEConv neighbor subtraction + sigmoid fitness
// =====================================================================
__global__ void pool_fit_kernel(const int* __restrict__ srcIdx, const int* __restrict__ alive,
                                const float* __restrict__ w3dot, const float* __restrict__ lecpart,
                                float* __restrict__ fit, float* __restrict__ fitm) {
  int n = blockIdx.x * blockDim.x + threadIdx.x;
  if (n >= NN) return;
  const int aD = alive[n];
  float s = 0.f;
#pragma unroll
  for (int r = 0; r < DEG; ++r) {
    const int sn = srcIdx[n * DEG + r];
    if (aD && alive[sn]) s += w3dot[sn];
  }
  const float lec = lecpart[n] - s;
  const float f = 1.f / (1.f + expf(-lec));
  fit[n] = f;
  fitm[n] = aD ? f : NEGF;
}

// =====================================================================
// pool stage 4: per-graph exact top-k (rank = #greater + #equal-with-smaller-idx,
// matching jax.lax.top_k stable tie-break) + xcur/alive update
// =====================================================================
__global__ void pool_topk_kernel(const float* __restrict__ fitm, const float* __restrict__ fit,
                                 const float* __restrict__ xc, int K, int* __restrict__ alive,
                                 float* __restrict__ xcur) {
  __shared__ float fv[NPG];
  const int b = blockIdx.x, tid = threadIdx.x;
  for (int r = tid; r < NPG; r += 256) fv[r] = fitm[b * NPG + r];
  __syncthreads();
  for (int r = tid; r < NPG; r += 256) {
    const float v = fv[r];
    int rank = 0;
    for (int j = 0; j < NPG; ++j) {
      const float u = fv[j];
      rank += (u > v) || (u == v && j < r);
    }
    const int n = b * NPG + r;
    const int na = (rank < K) && alive[n];
    alive[n] = na;
    const float fr = fit[n];
    if (na) {
      for (int f = 0; f < HH; ++f) xcur[(size_t)n * HH + f] = xc[(size_t)n * HH + f] * fr;
    } else {
      for (int f = 0; f < HH; ++f) xcur[(size_t)n * HH + f] = 0.f;
    }
  }
}

// =====================================================================
// head: xcat(10*64) -> lin1 relu -> lin2 -> log_softmax. one block / graph
// =====================================================================
__global__ void head_kernel(const float* __restrict__ xs, const float* __restrict__ w1,
                            const float* __restrict__ b1, const float* __restrict__ w2,
                            const float* __restrict__ b2, float* __restrict__ out) {
  __shared__ float o1[HH];
  __shared__ float o2[NCLS];
  const int b = blockIdx.x, j = threadIdx.x;
  float acc = b1[j];
  for (int s = 0; s < 10; ++s)
    for (int f = 0; f < HH; ++f)
      acc += xs[(size_t)s * NB * HH + b * HH + f] * w1[(s * HH + f) * HH + j];
  o1[j] = fmaxf(acc, 0.f);
  __syncthreads();
  if (j < NCLS) {
    float a2 = b2[j];
    for (int f = 0; f < HH; ++f) a2 += o1[f] * w2[f * NCLS + j];
    o2[j] = a2;
  }
  __syncthreads();
  if (j == 0) {
    float mx = o2[0];
    for (int c = 1; c < NCLS; ++c) mx = fmaxf(mx, o2[c]);
    float s = 0.f;
    for (int c = 0; c < NCLS; ++c) s += expf(o2[c] - mx);
    const float lse = mx + logf(s);
    for (int c = 0; c < NCLS; ++c) out[b * NCLS + c] = o2[c] - lse;
  }
}

// =====================================================================
extern "C" void kernel_launch(void* const* d_in, const int* in_sizes, int n_in,
                              void* d_out, int out_size, void* d_ws, size_t ws_size,
                              hipStream_t stream) {
  (void)in_sizes; (void)n_in; (void)out_size; (void)ws_size;
  const float* x      = (const float*)d_in[0];
  const float* pos    = (const float*)d_in[1];
  const float* eattr  = (const float*)d_in[2];
  const float* pew    = (const float*)d_in[3];
  const float* peb    = (const float*)d_in[4];
  const float* prw    = (const float*)d_in[5];
  const float* prb    = (const float*)d_in[6];
  const float* ppw    = (const float*)d_in[7];
  const float* ppb    = (const float*)d_in[8];
  const float* plw    = (const float*)d_in[9];
  const float* plb    = (const float*)d_in[10];
  const float* ecw0   = (const float*)d_in[11];
  const float* ecb0   = (const float*)d_in[12];
  const float* ecw1   = (const float*)d_in[13];
  const float* ecb1   = (const float*)d_in[14];
  const float* ecw2   = (const float*)d_in[15];
  const float* ecb2   = (const float*)d_in[16];
  const float* poolw  = (const float*)d_in[17];
  const float* poolb  = (const float*)d_in[18];
  const float* patt   = (const float*)d_in[19];
  const float* lew1   = (const float*)d_in[20];
  const float* lew2   = (const float*)d_in[21];
  const float* lew3   = (const float*)d_in[22];
  const float* leb    = (const float*)d_in[23];
  const float* l1w    = (const float*)d_in[24];
  const float* l1b    = (const float*)d_in[25];
  const float* l2w    = (const float*)d_in[26];
  const float* l2b    = (const float*)d_in[27];
  const int*   eidx   = (const int*)d_in[28];   // [2][E]; row0 = src (dst == e/8)
  const int*   srcIdx = eidx;

  // workspace layout
  float* ws = (float*)d_ws;
  size_t o = 0;
  float* xcur   = ws + o; o += (size_t)NN * HH;
  float* baseB  = ws + o; o += (size_t)NN * HH;
  float* xcB    = ws + o; o += (size_t)NN * HH;
  float* qdotB  = ws + o; o += NN;
  float* xdotB  = ws + o; o += NN;
  float* ssB    = ws + o; o += NN;
  float* w3dotB = ws + o; o += NN;
  float* lecB   = ws + o; o += NN;
  float* fitB   = ws + o; o += NN;
  float* fitmB  = ws + o; o += NN;
  int*   alive  = (int*)(ws + o); o += NN;
  float* xsB    = ws + o; o += (size_t)10 * NB * HH;

  static const int KS[4] = {1434, 1004, 703, 493};

  init_alive_kernel<<<NN / 256, 256, 0, stream>>>(alive);
  pna_kernel<<<NN / 256, 256, 0, stream>>>(x, pos, eattr, pew, peb, prw, prb, ppw, ppb,
                                           plw, plb, srcIdx, xcur);
  gmp_kernel<<<NB, 256, 0, stream>>>(xcur, alive, xsB);

  for (int i = 0; i < NLAY; ++i) {
    const float* w0 = ecw0 + (size_t)i * 70 * 64;
    base_kernel<<<BASE_BLOCKS, 128, 0, stream>>>(xcur, pos, w0, ecb0 + i * 64, baseB);
    edge_kernel<<<EDGE_BLOCKS, 128, 0, stream>>>(baseB, pos, srcIdx, alive, w0,
                                                 ecb1 + i * 64, ecw1 + (size_t)i * 4096,
                                                 ecb2 + i * 64, ecw2 + (size_t)i * 4096, xcur);
    gmp_kernel<<<NB, 256, 0, stream>>>(xcur, alive, xsB + (size_t)(i + 1) * NB * HH);
    if (i % 2 == 0 && i < NLAY - 1) {
      const int p = i / 2;
      pool_q_kernel<<<NN / 256, 256, 0, stream>>>(xcur, srcIdx, alive,
                                                  poolw + (size_t)p * 4096, poolb + p * 64,
                                                  patt + p * 128, qdotB, xdotB, ssB);
      pool_xc_kernel<<<NN / 256, 256, 0, stream>>>(xcur, srcIdx, alive, qdotB, xdotB, ssB,
                                                   lew1 + p * 64, lew2 + p * 64,
                                                   lew3 + p * 64, leb + p, xcB, w3dotB, lecB);
      pool_fit_kernel<<<NN / 256, 256, 0, stream>>>(srcIdx, alive, w3dotB, lecB, fitB, fitmB);
      pool_topk_kernel<<<NB, 256, 0, stream>>>(fitmB, fitB, xcB, KS[p], alive, xcur);
    }
  }

  head_kernel<<<NB, 64, 0, stream>>>(xsB, l1w, l1b, l2w, l2b, (float*)d_out);
}